// CTLN_49314814493112
// MI455X (gfx1250) — compile-verified
//
#include <hip/hip_runtime.h>
#include <stdint.h>

// CTLN recurrence on MI455X (gfx1250): persistent-block iterated GEMV.
//   x_{t+1} = (1-dt)*x_t + dt*relu(W@x_t + theta + u[:,t])
// 64 persistent blocks; each owns 32 rows of W, kept in LDS (256KB+pad) for
// all 8192 steps. Inner product on the matrix pipe via V_WMMA_F32_16X16X4_F32
// (exact f32). Per-step state exchange through L2 with DEV-scope atomics and
// a slot-per-step grid barrier held in d_ws (replay-deterministic: slots are
// re-zeroed under a flag handshake at the start of every call, flags cleared
// at the end).

#define N      2048
#define T      8192
#define NB     64            // persistent blocks (assumes >= 64 WGPs resident)
#define BT     512           // threads/block = 16 wave32
#define ROWS   32            // rows of W per block
#define PITCH  2052          // LDS row pitch (floats): 4-bank shift/row -> conflict-free
#define DT     0.1f
#define MAGICF 0x1234ABCDu

typedef float v2f __attribute__((ext_vector_type(2)));
typedef float v4f __attribute__((ext_vector_type(4)));
typedef float v8f __attribute__((ext_vector_type(8)));

extern __shared__ float smem[];

__global__ __launch_bounds__(BT, 1)
void ctln_persistent(const float* __restrict__ x0,
                     const float* __restrict__ u,
                     const float* __restrict__ W,
                     const float* __restrict__ theta,
                     float* __restrict__ out,
                     uint32_t* __restrict__ slots,   // T barrier counters
                     uint32_t* __restrict__ flags,   // NB init flags
                     float* __restrict__ xcur)       // N-float shared state
{
  float* WLds = smem;                       // ROWS * PITCH floats
  float* xLds = smem + ROWS * PITCH;        // N floats (full state vector)
  float* pLds = xLds + N;                   // 16 waves * 16 partials

  const int tid    = threadIdx.x;
  const int bid    = blockIdx.x;
  const int lane   = tid & 31;              // wave32
  const int wid    = tid >> 5;              // 0..15
  const int tileId = wid >> 3;              // 0..1  (16-row tile)
  const int kPart  = wid & 7;               // 0..7  (256-wide K chunk)
  const int hi     = lane >> 4;             // lane group selects K pair {0,1}/{2,3}
  const int rIn    = lane & 15;
  const int rbase  = bid * ROWS;

  // ---- init: zero this call's barrier slots, publish flag, stage W, wait all
  for (int i = bid * BT + tid; i < T; i += NB * BT)
    __hip_atomic_store(&slots[i], 0u, __ATOMIC_RELAXED, __HIP_MEMORY_SCOPE_AGENT);
  __syncthreads();
  if (tid == 0)
    __hip_atomic_store(&flags[bid], MAGICF, __ATOMIC_RELEASE, __HIP_MEMORY_SCOPE_AGENT);

  // Stage W slice (ROWS x N) into LDS with padded pitch (streamed, last-use).
  for (int idx = tid; idx < ROWS * (N / 4); idx += BT) {
    const int r  = idx / (N / 4);
    const int c4 = (idx % (N / 4)) * 4;
    const v4f v =
        __builtin_nontemporal_load((const v4f*)&W[(size_t)(rbase + r) * N + c4]);
    *(v4f*)&WLds[r * PITCH + c4] = v;
  }
  // x for step 0 comes straight from x0 (no reliance on ws contents).
  for (int i = tid; i < N; i += BT) xLds[i] = x0[i];

  if (tid < NB) {
    while (__hip_atomic_load(&flags[tid], __ATOMIC_ACQUIRE, __HIP_MEMORY_SCOPE_AGENT)
           != MAGICF)
      __builtin_amdgcn_s_sleep(1);
  }
  __syncthreads();

  const float thRow = (tid < ROWS) ? theta[rbase + tid] : 0.0f;
  const float* wRow = &WLds[(tileId * 16 + rIn) * PITCH];

  for (int t = 0; t < T; ++t) {
    // Prefetch u cacheline ~16 steps ahead (one 64B line = 16 consecutive t).
    if (tid < ROWS) {
      int tp = t + 16; if (tp > T - 1) tp = T - 1;
      __builtin_prefetch(&u[(size_t)(rbase + tid) * T + tp], 0, 1);
    }

    // ---- GEMV partial on the matrix pipe.
    // D = A*B + C with A[m,k'] = x[k0+k'] (broadcast rows),
    // B[k',n] = W[rbase + tileId*16 + n][k0+k'].  Every row of D holds the
    // same 16 partial dot products -> c[0], lanes 0..15 is the result.
    // A frag: lanes<16 want K{0,1}, lanes>=16 K{2,3}  -> float2 at k0+2*hi.
    // B frag: lane n reads W row (tile*16 + n), same K pair -> float2.
    v8f c0 = {}; v8f c1 = {};
    const int kb = kPart * 256;
    #pragma unroll 4
    for (int k0 = 0; k0 < 128; k0 += 4) {
      {
        v2f a = *(const v2f*)&xLds[kb + k0 + 2 * hi];
        v2f b = *(const v2f*)&wRow[kb + k0 + 2 * hi];
        c0 = __builtin_amdgcn_wmma_f32_16x16x4_f32(false, a, false, b,
                                                   (short)0, c0, false, false);
      }
      {
        v2f a = *(const v2f*)&xLds[kb + 128 + k0 + 2 * hi];
        v2f b = *(const v2f*)&wRow[kb + 128 + k0 + 2 * hi];
        c1 = __builtin_amdgcn_wmma_f32_16x16x4_f32(false, a, false, b,
                                                   (short)0, c1, false, false);
      }
    }
    if (lane < 16) pLds[wid * 16 + rIn] = c0[0] + c1[0];
    __syncthreads();

    // ---- reduce 8 K-partials + pointwise update for the 32 owned rows
    if (tid < ROWS) {
      const int tl = tid >> 4, rr = tid & 15;
      float y = 0.0f;
      #pragma unroll
      for (int p = 0; p < 8; ++p) y += pLds[(tl * 8 + p) * 16 + rr];
      const int   row  = rbase + tid;
      const float uv   = u[(size_t)row * T + t];
      float pre = y + thRow + uv;
      pre = pre > 0.0f ? pre : 0.0f;
      const float xnew = (1.0f - DT) * xLds[row] + DT * pre;
      out[(size_t)row * T + t] = xnew;
      __hip_atomic_store(&xcur[row], xnew, __ATOMIC_RELAXED, __HIP_MEMORY_SCOPE_AGENT);
      __threadfence();   // make DEV-scope state stores visible before arrive
    }
    __syncthreads();

    // ---- grid barrier (dedicated pre-zeroed slot per step)
    if (tid == 0) {
      __hip_atomic_fetch_add(&slots[t], 1u, __ATOMIC_ACQ_REL, __HIP_MEMORY_SCOPE_AGENT);
      while (__hip_atomic_load(&slots[t], __ATOMIC_ACQUIRE, __HIP_MEMORY_SCOPE_AGENT) < NB)
        __builtin_amdgcn_s_sleep(0);
    }
    __syncthreads();

    // ---- broadcast x_{t+1} (DEV-scope loads force-miss WGP$ -> L2-fresh)
    if (t + 1 < T) {
      for (int i = tid; i < N; i += BT)
        xLds[i] = __hip_atomic_load(&xcur[i], __ATOMIC_RELAXED, __HIP_MEMORY_SCOPE_AGENT);
      __syncthreads();
    }
  }

  // Leave ws in a deterministic state for the next replay: flags must be 0.
  if (tid == 0)
    __hip_atomic_store(&flags[bid], 0u, __ATOMIC_RELEASE, __HIP_MEMORY_SCOPE_AGENT);
}

extern "C" void kernel_launch(void* const* d_in, const int* in_sizes, int n_in,
                              void* d_out, int out_size, void* d_ws, size_t ws_size,
                              hipStream_t stream) {
  (void)in_sizes; (void)n_in; (void)out_size; (void)ws_size;
  const float* x0 = (const float*)d_in[0];
  const float* u  = (const float*)d_in[1];
  const float* W  = (const float*)d_in[2];
  const float* th = (const float*)d_in[3];
  float* out = (float*)d_out;

  // ws layout: [T barrier slots][NB flags][N-float state]  (~41 KB)
  uint32_t* slots = (uint32_t*)d_ws;
  uint32_t* flags = slots + T;
  float*    xcur  = (float*)(flags + NB);

  const size_t shmem = (size_t)(ROWS * PITCH + N + 16 * 16) * sizeof(float); // ~272 KB
  (void)hipFuncSetAttribute((const void*)ctln_persistent,
                            hipFuncAttributeMaxDynamicSharedMemorySize, (int)shmem);
  hipLaunchKernelGGL(ctln_persistent, dim3(NB), dim3(BT), shmem, stream,
                     x0, u, W, th, out, slots, flags, xcur);
}